// TopKGate_11982958756385
// MI455X (gfx1250) — compile-verified
//
#include <hip/hip_runtime.h>
#include <hip/hip_bf16.h>

typedef __attribute__((ext_vector_type(2))) float v2f;
typedef __attribute__((ext_vector_type(8))) float v8f;

#define S_TOK   8192
#define N_EXP   64
#define MDIM    2048
#define CAP     128

// ---------------------------------------------------------------------------
// Kernel 1: logits[8192,64] = x[8192,2048] @ wg[64,2048]^T via WMMA f32 16x16x4.
// Block = 128 threads = 4 waves; block owns a 16-row stripe, wave w owns
// experts [16w, 16w+16). A 16x4 f32 tile: lanes 0-15 -> K=0,1; lanes 16-31 ->
// K=2,3 (2 VGPRs). B mirrored. Both map to float2 global loads.
// ---------------------------------------------------------------------------
__global__ void __launch_bounds__(128) k_gemm_logits(const float* __restrict__ x,
                                                     const float* __restrict__ wg,
                                                     float* __restrict__ logits) {
    const int wave = threadIdx.x >> 5;
    const int lane = threadIdx.x & 31;
    const int half = lane >> 4;     // 0: K=0,1   1: K=2,3
    const int lrow = lane & 15;
    const int m0 = blockIdx.x * 16;
    const int nbase = wave * 16;

    const float* arow = x  + (size_t)(m0 + lrow)    * MDIM + half * 2;
    const float* brow = wg + (size_t)(nbase + lrow) * MDIM + half * 2;

    v8f acc = {};
#pragma unroll 4
    for (int k = 0; k < MDIM; k += 4) {
        v2f a = *(const v2f*)(arow + k);
        v2f b = *(const v2f*)(brow + k);
        acc = __builtin_amdgcn_wmma_f32_16x16x4_f32(
            /*neg_a=*/false, a, /*neg_b=*/false, b,
            /*c_mod=*/(short)0, acc, /*reuse_a=*/false, /*reuse_b=*/false);
    }

    // C/D layout: VGPR i -> lanes 0-15: M=i, lanes 16-31: M=8+i; N=lane%16
    float* out = logits + (size_t)(m0 + half * 8) * N_EXP + nbase + lrow;
#pragma unroll
    for (int i = 0; i < 8; ++i) out[(size_t)i * N_EXP] = acc[i];
}

// ---------------------------------------------------------------------------
// Kernel 2: per-token softmax/argmax + deterministic per-block expert partials
// (gate-sum & histogram) + intra-block rank (token-order position in expert
// queue). 32 blocks x 256 tokens.
// ---------------------------------------------------------------------------
__global__ void __launch_bounds__(256) k_softmax_rank(const float* __restrict__ logits,
                                                      int* __restrict__ expert,
                                                      float* __restrict__ gate,
                                                      int* __restrict__ rank,
                                                      int* __restrict__ hist,
                                                      float* __restrict__ gpart) {
    __shared__ float sm[256];    // row max
    __shared__ float sinv[256];  // 1/denominator
    __shared__ int   sex[256];   // argmax expert
    __shared__ float gp[256];    // quarter partial gate sums (q*64+e)
    __shared__ int   cp[256];    // quarter partial counts

    const int t  = threadIdx.x;
    const int s0 = blockIdx.x * 256;
    const int s  = s0 + t;
    const float* row = logits + (size_t)s * N_EXP;

    float m = -3.4e38f; int am = 0;
#pragma unroll
    for (int e = 0; e < N_EXP; ++e) { float v = row[e]; if (v > m) { m = v; am = e; } }
    float d = 0.f;
#pragma unroll
    for (int e = 0; e < N_EXP; ++e) d += __expf(row[e] - m);
    const float inv = 1.0f / d;          // == softmax value at argmax

    sm[t] = m; sinv[t] = inv; sex[t] = am;
    expert[s] = am;
    gate[s]   = inv;
    __syncthreads();

    // deterministic per-expert partials: thread (q,e) sums 64 tokens in order
    const int q = t >> 6, e = t & 63;
    float gs = 0.f; int cs = 0;
    for (int i = 0; i < 64; ++i) {
        const int tt = q * 64 + i;
        gs += __expf(logits[(size_t)(s0 + tt) * N_EXP + e] - sm[tt]) * sinv[tt];
        cs += (sex[tt] == e) ? 1 : 0;
    }
    gp[t] = gs; cp[t] = cs;

    // intra-block rank: # earlier tokens in this block with same expert
    int r = 0;
    for (int i = 0; i < t; ++i) r += (sex[i] == am) ? 1 : 0;
    rank[s] = r;
    __syncthreads();

    if (t < 64) {
        hist [blockIdx.x * 64 + t] = cp[t] + cp[64 + t] + cp[128 + t] + cp[192 + t];
        gpart[blockIdx.x * 64 + t] = gp[t] + gp[64 + t] + gp[128 + t] + gp[192 + t];
    }
}

// ---------------------------------------------------------------------------
// Kernel 3: exclusive prefix over the 32 block histograms -> per-block queue
// offsets, plus final per-expert totals (count, gate-sum). 1 block, 64 thr.
// ---------------------------------------------------------------------------
__global__ void __launch_bounds__(64) k_prefix(const int* __restrict__ hist,
                                               const float* __restrict__ gpart,
                                               int* __restrict__ offs,
                                               float* __restrict__ gsum,
                                               int* __restrict__ cnt) {
    const int e = threadIdx.x;
    int run = 0; float gs = 0.f;
    for (int b = 0; b < 32; ++b) {
        offs[b * 64 + e] = run;
        run += hist[b * 64 + e];
        gs  += gpart[b * 64 + e];
    }
    cnt[e]  = run;
    gsum[e] = gs;
}

// ---------------------------------------------------------------------------
// Kernel 4: zero-fill entire d_out (combine_weights + dispatch_mask are
// ~0.012% dense; d_out is poisoned). float4 grid-stride stores.
// ---------------------------------------------------------------------------
__global__ void __launch_bounds__(256) k_zero(float* __restrict__ out, long long n) {
    const long long stride = (long long)gridDim.x * blockDim.x;
    long long i = (long long)blockIdx.x * blockDim.x + threadIdx.x;
    const long long nv = n >> 2;
    float4 z = make_float4(0.f, 0.f, 0.f, 0.f);
    float4* o4 = (float4*)out;
    for (long long v = i; v < nv; v += stride) o4[v] = z;
    if (i < (n & 3)) out[nv * 4 + i] = 0.f;
}

// ---------------------------------------------------------------------------
// Kernel 5: scatter the <=8192 nonzeros + l_aux.
// loc = block_offset[expert] + intra_block_rank; drop if loc >= capacity.
// l_aux = mean(me*ce)*E^2 = 64 * sum_e(gsum_e * cnt_e) / 8192^2.
// ---------------------------------------------------------------------------
__global__ void __launch_bounds__(256) k_scatter(const int* __restrict__ expert,
                                                 const float* __restrict__ gate,
                                                 const int* __restrict__ rank,
                                                 const int* __restrict__ offs,
                                                 const float* __restrict__ gsum,
                                                 const int* __restrict__ cnt,
                                                 float* __restrict__ d_out) {
    const int s = blockIdx.x * 256 + threadIdx.x;
    float* cw = d_out + 1;
    float* dm = d_out + 1 + (size_t)S_TOK * N_EXP * CAP;

    const int e   = expert[s];
    const int loc = offs[(s >> 8) * 64 + e] + rank[s];
    if (loc < CAP) {
        const size_t base = (size_t)s * (N_EXP * CAP) + (size_t)e * CAP + loc;
        cw[base] = gate[s];
        dm[base] = 1.0f;
    }

    if (blockIdx.x == 0) {
        __shared__ float terms[64];
        const int t = threadIdx.x;
        if (t < 64) terms[t] = gsum[t] * (float)cnt[t];
        __syncthreads();
        if (t == 0) {
            float tot = 0.f;
            for (int i = 0; i < 64; ++i) tot += terms[i];
            d_out[0] = tot * (64.0f / (8192.0f * 8192.0f));
        }
    }
}

// ---------------------------------------------------------------------------
extern "C" void kernel_launch(void* const* d_in, const int* in_sizes, int n_in,
                              void* d_out, int out_size, void* d_ws, size_t ws_size,
                              hipStream_t stream) {
    const float* x  = (const float*)d_in[0];   // [8192, 2048]
    const float* wg = (const float*)d_in[1];   // [64, 2048]
    float* out = (float*)d_out;

    char* ws = (char*)d_ws;
    float* logits = (float*)(ws);                              // 8192*64*4 = 2 MB
    int*   expert = (int*)  (ws + 2097152);                    // 32 KB
    float* gate   = (float*)(ws + 2097152 + 32768);            // 32 KB
    int*   rank   = (int*)  (ws + 2097152 + 65536);            // 32 KB
    int*   hist   = (int*)  (ws + 2097152 + 98304);            // 32*64*4 = 8 KB
    int*   offs   = (int*)  (ws + 2097152 + 106496);           // 8 KB
    float* gpart  = (float*)(ws + 2097152 + 114688);           // 8 KB
    float* gsum   = (float*)(ws + 2097152 + 122880);           // 256 B
    int*   cnt    = (int*)  (ws + 2097152 + 123136);           // 256 B

    k_gemm_logits<<<S_TOK / 16, 128, 0, stream>>>(x, wg, logits);
    k_softmax_rank<<<S_TOK / 256, 256, 0, stream>>>(logits, expert, gate, rank, hist, gpart);
    k_prefix<<<1, 64, 0, stream>>>(hist, gpart, offs, gsum, cnt);
    k_zero<<<4096, 256, 0, stream>>>(out, (long long)out_size);
    k_scatter<<<S_TOK / 256, 256, 0, stream>>>(expert, gate, rank, offs, gsum, cnt, out);
}